// TemporalGraphNetwork_9663676416704
// MI455X (gfx1250) — compile-verified
//
#include <hip/hip_runtime.h>

#define N_NODES 50000
#define N_EDGES 800000
#define E_TOT   (N_EDGES + N_NODES)
#define MEM_N   1000
#define HID     256

typedef float v2f __attribute__((ext_vector_type(2)));
typedef float v8f __attribute__((ext_vector_type(8)));

// ---------------------------------------------------------------------------
// WMMA f32 GEMM: C[M,N] = A[M,K] @ B[K,N]  (+bias) (+=C) (relu)
// Block: 256 threads = 8 wave32. 64x64 tile per block; each wave owns one
// 16x16 m-tile x two 16x16 n-tiles. K blocked by 32 through LDS.
//  - Full interior tiles staged with GLOBAL_LOAD_ASYNC_TO_LDS_B32 (ASYNCcnt),
//    including transpose-on-store for the B tile (per-lane LDS destinations).
//  - Edge tiles fall back to guarded scalar loads with zero-fill (handles
//    M%64 != 0 and K=257).
//  - Both A and B^T tiles padded to 34 floats/row: every WMMA fragment is a
//    contiguous 8B LDS read on conflict-free banks.
// ---------------------------------------------------------------------------
__global__ __launch_bounds__(256) void gemm_wmma_f32(
    const float* __restrict__ A, const float* __restrict__ B,
    const float* __restrict__ bias, float* __restrict__ C,
    int M, int N, int K, int flags)
{
  __shared__ float As[64][34];   // A tile  [m][k]
  __shared__ float Bs[64][34];   // B^T tile [n][k]

  const int tid  = threadIdx.x;
  const int lane = tid & 31;
  const int wv   = tid >> 5;
  const int tm   = wv & 3;              // m-tile 0..3
  const int tnb  = (wv >> 2) << 1;      // first of two n-tiles {0,2}
  const int m0   = blockIdx.y << 6;
  const int n0   = blockIdx.x << 6;
  const int mloc = lane & 15;
  const int koff = (lane >> 4) << 1;    // lanes 16..31 hold K+2,K+3 halves

  const unsigned asBase = (unsigned)(size_t)&As[0][0];
  const unsigned bsBase = (unsigned)(size_t)&Bs[0][0];

  v8f acc0 = {0.f,0.f,0.f,0.f,0.f,0.f,0.f,0.f};
  v8f acc1 = {0.f,0.f,0.f,0.f,0.f,0.f,0.f,0.f};

  const bool full_m = (m0 + 64 <= M);   // block-uniform

  for (int kb = 0; kb < K; kb += 32) {
    if (full_m && (kb + 32 <= K)) {
      // ---- async global->LDS staging (no VGPR round-trip) ----
#pragma unroll
      for (int i = 0; i < 8; ++i) {     // A tile 64x32, row-major
        int idx = (i << 8) + tid;
        int r = idx >> 5, c = idx & 31;
        unsigned loff = asBase + ((unsigned)(r * 34 + c) << 2);
        unsigned goff = (((unsigned)(m0 + r) * (unsigned)K) + (unsigned)(kb + c)) << 2;
        asm volatile("global_load_async_to_lds_b32 %0, %1, %2"
                     :: "v"(loff), "v"(goff), "s"(A) : "memory");
      }
#pragma unroll
      for (int i = 0; i < 8; ++i) {     // B tile 32x64, transposed on store
        int idx = (i << 8) + tid;
        int k = idx >> 6, n = idx & 63; // consecutive lanes -> consecutive n (coalesced)
        unsigned loff = bsBase + ((unsigned)(n * 34 + k) << 2);
        unsigned goff = (((unsigned)(kb + k) * (unsigned)N) + (unsigned)(n0 + n)) << 2;
        asm volatile("global_load_async_to_lds_b32 %0, %1, %2"
                     :: "v"(loff), "v"(goff), "s"(B) : "memory");
      }
      asm volatile("s_wait_asynccnt 0" ::: "memory");
    } else {
      // ---- guarded fallback with zero-fill (edge tiles / K=257) ----
#pragma unroll
      for (int i = 0; i < 8; ++i) {
        int idx = (i << 8) + tid;
        int r = idx >> 5, c = idx & 31;
        int gm = m0 + r, gk = kb + c;
        As[r][c] = (gm < M && gk < K) ? A[(size_t)gm * K + gk] : 0.f;
      }
#pragma unroll
      for (int i = 0; i < 8; ++i) {
        int idx = (i << 8) + tid;
        int k = idx >> 6, n = idx & 63;
        int gk = kb + k, gn = n0 + n;
        Bs[n][k] = (gk < K && gn < N) ? B[(size_t)gk * N + gn] : 0.f;
      }
    }
    __syncthreads();
#pragma unroll
    for (int kk = 0; kk < 32; kk += 4) {
      v2f a  = *(const v2f*)&As[(tm << 4) + mloc][kk + koff];
      v2f b0 = *(const v2f*)&Bs[(tnb << 4) + mloc][kk + koff];
      v2f b1 = *(const v2f*)&Bs[(tnb << 4) + 16 + mloc][kk + koff];
      acc0 = __builtin_amdgcn_wmma_f32_16x16x4_f32(false, a, false, b0,
                                                   (short)0, acc0, false, false);
      acc1 = __builtin_amdgcn_wmma_f32_16x16x4_f32(false, a, false, b1,
                                                   (short)0, acc1, false, false);
    }
    __syncthreads();
  }

  const bool do_relu  = (flags & 1) != 0;
  const bool do_accum = (flags & 2) != 0;
#pragma unroll
  for (int r = 0; r < 8; ++r) {
    int m = m0 + (tm << 4) + r + ((lane >> 4) << 3);
    if (m < M) {
      int n = n0 + (tnb << 4) + mloc;
      float v0 = acc0[r];
      float v1 = acc1[r];
      if (bias) { v0 += bias[n]; v1 += bias[n + 16]; }
      size_t p0 = (size_t)m * N + n;
      if (do_accum) { v0 += C[p0]; v1 += C[p0 + 16]; }
      if (do_relu)  { v0 = fmaxf(v0, 0.f); v1 = fmaxf(v1, 0.f); }
      C[p0]      = v0;
      C[p0 + 16] = v1;
    }
  }
}

// ---------------------------------------------------------------------------
// Small utility kernels
// ---------------------------------------------------------------------------
__global__ void fill_f32(float* p, float v, long long n) {
  long long i = (long long)blockIdx.x * blockDim.x + threadIdx.x;
  long long st = (long long)gridDim.x * blockDim.x;
  for (; i < n; i += st) p[i] = v;
}

__global__ void relu_ip(float* p, long long n) {
  long long i = (long long)blockIdx.x * blockDim.x + threadIdx.x;
  long long st = (long long)gridDim.x * blockDim.x;
  for (; i < n; i += st) p[i] = fmaxf(p[i], 0.f);
}

__global__ void init_bias_rows(float* out, const float* __restrict__ bias, int n_rows) {
  long long total = (long long)n_rows * HID;
  long long i = (long long)blockIdx.x * blockDim.x + threadIdx.x;
  long long st = (long long)gridDim.x * blockDim.x;
  for (; i < total; i += st) out[i] = bias[i & (HID - 1)];
}

__global__ void reduce_sum(const float* __restrict__ x, int n, float* out) {
  __shared__ float s[256];
  float v = 0.f;
  for (int i = blockIdx.x * blockDim.x + threadIdx.x; i < n;
       i += gridDim.x * blockDim.x) v += x[i];
  s[threadIdx.x] = v;
  __syncthreads();
  for (int st = 128; st > 0; st >>= 1) {
    if (threadIdx.x < st) s[threadIdx.x] += s[threadIdx.x + st];
    __syncthreads();
  }
  if (threadIdx.x == 0) atomicAdd(out, s[0]);
}

// mem_in = concat(node_memory[1000,256], t_mean - memory_time[1000,1]) -> [1000,257]
__global__ void build_memin(const float* __restrict__ node_memory,
                            const float* __restrict__ memory_time,
                            const float* __restrict__ ts_sum,
                            float* __restrict__ memin) {
  int i = blockIdx.x * blockDim.x + threadIdx.x;
  if (i >= MEM_N * 257) return;
  int r = i / 257, c = i % 257;
  float t_mean = ts_sum[0] * (1.0f / (float)N_NODES);
  memin[i] = (c < HID) ? node_memory[r * HID + c] : (t_mean - memory_time[r]);
}

// es/ed per (node, head): dot of h[node, head*F : head*F+F] with a_src/a_dst rows
__global__ void attn_scores(const float* __restrict__ hfeat,
                            const float* __restrict__ a_src,
                            const float* __restrict__ a_dst,
                            float* __restrict__ es, float* __restrict__ ed,
                            int n, int H, int F) {
  int i = blockIdx.x * blockDim.x + threadIdx.x;
  if (i >= n * H) return;
  int node = i / H, head = i % H;
  const float* hp = hfeat + (size_t)node * HID + head * F;
  const float* as = a_src + head * F;
  const float* ad = a_dst + head * F;
  float s = 0.f, d = 0.f;
  for (int f = 0; f < F; ++f) {
    float hv = hp[f];
    s += hv * as[f];
    d += hv * ad[f];
  }
  es[i] = s;
  ed[i] = d;
}

// sign-magnitude float atomic max (portable; converges for mixed signs, init -inf)
__device__ __forceinline__ void atomicMaxF(float* addr, float v) {
  if (v >= 0.f) atomicMax((int*)addr, __float_as_int(v));
  else          atomicMin((unsigned int*)addr, __float_as_uint(v));
}

__device__ __forceinline__ void edge_endpoints(const long long* __restrict__ ei,
                                               int e, int* src, int* dst) {
  if (e < N_EDGES) { *src = (int)ei[e]; *dst = (int)ei[N_EDGES + e]; }
  else             { *src = *dst = e - N_EDGES; }  // appended self-loops
}

// pass 1: e = leaky_relu(es[src]+ed[dst], 0.2); stash in wbuf; segment-max into m
__global__ void edge_max(const long long* __restrict__ ei,
                         const float* __restrict__ es, const float* __restrict__ ed,
                         float* __restrict__ wbuf, float* __restrict__ m, int H) {
  int i = blockIdx.x * blockDim.x + threadIdx.x;
  if (i >= E_TOT * H) return;
  int e = i / H, h = i % H;
  int src, dst;
  edge_endpoints(ei, e, &src, &dst);
  float v = es[src * H + h] + ed[dst * H + h];
  v = (v > 0.f) ? v : 0.2f * v;
  wbuf[i] = v;
  atomicMaxF(&m[dst * H + h], v);
}

// pass 2: w = exp(e - m[dst]); segment-sum into z
__global__ void edge_expsum(const long long* __restrict__ ei,
                            float* __restrict__ wbuf, const float* __restrict__ m,
                            float* __restrict__ z, int H) {
  int i = blockIdx.x * blockDim.x + threadIdx.x;
  if (i >= E_TOT * H) return;
  int e = i / H, h = i % H;
  int src, dst;
  edge_endpoints(ei, e, &src, &dst);
  float v = __expf(wbuf[i] - m[dst * H + h]);
  wbuf[i] = v;
  atomicAdd(&z[dst * H + h], v);
}

// pass 3: out[dst] += (w/z[dst]) * h[src]   (one 256-thread block per edge)
__global__ __launch_bounds__(256) void edge_aggregate(
    const long long* __restrict__ ei, const float* __restrict__ hfeat,
    const float* __restrict__ wbuf, const float* __restrict__ z,
    float* __restrict__ out, int H) {
  int e = blockIdx.x;
  int c = threadIdx.x;
  int src, dst;
  edge_endpoints(ei, e, &src, &dst);
  int F = HID / H;
  int head = c / F;
  float alpha = wbuf[(size_t)e * H + head] / z[dst * H + head];
  atomicAdd(&out[(size_t)dst * HID + c], alpha * hfeat[(size_t)src * HID + c]);
}

// logits: one wave32 per node; 2-wide output via shuffle reduction
__global__ __launch_bounds__(256) void logits_kernel(
    const float* __restrict__ hfeat, const float* __restrict__ w,
    const float* __restrict__ b, float* __restrict__ out, int n) {
  int wave = threadIdx.x >> 5, lane = threadIdx.x & 31;
  int node = blockIdx.x * 8 + wave;
  if (node >= n) return;
  float a0 = 0.f, a1 = 0.f;
  for (int c = lane; c < HID; c += 32) {
    float hv = hfeat[(size_t)node * HID + c];
    a0 += hv * w[c * 2 + 0];
    a1 += hv * w[c * 2 + 1];
  }
  for (int off = 16; off > 0; off >>= 1) {
    a0 += __shfl_down(a0, off, 32);
    a1 += __shfl_down(a1, off, 32);
  }
  if (lane == 0) {
    out[node * 2 + 0] = a0 + b[0];
    out[node * 2 + 1] = a1 + b[1];
  }
}

// ---------------------------------------------------------------------------
extern "C" void kernel_launch(void* const* d_in, const int* in_sizes, int n_in,
                              void* d_out, int out_size, void* d_ws, size_t ws_size,
                              hipStream_t stream) {
  const float*     x        = (const float*)d_in[0];
  const long long* ei       = (const long long*)d_in[1];
  const float*     ts       = (const float*)d_in[2];
  const float*     node_mem = (const float*)d_in[3];
  const float*     mem_time = (const float*)d_in[4];
  const float*     te_w1    = (const float*)d_in[5];
  const float*     te_b1    = (const float*)d_in[6];
  const float*     te_w2    = (const float*)d_in[7];
  const float*     te_b2    = (const float*)d_in[8];
  const float*     g1_w     = (const float*)d_in[9];
  const float*     g1_asrc  = (const float*)d_in[10];
  const float*     g1_adst  = (const float*)d_in[11];
  const float*     g1_b     = (const float*)d_in[12];
  const float*     g2_w     = (const float*)d_in[13];
  const float*     g2_asrc  = (const float*)d_in[14];
  const float*     g2_adst  = (const float*)d_in[15];
  const float*     g2_b     = (const float*)d_in[16];
  const float*     c_w1     = (const float*)d_in[17];
  const float*     c_b1     = (const float*)d_in[18];
  const float*     c_w2     = (const float*)d_in[19];
  const float*     c_b2     = (const float*)d_in[20];

  float* ws = (float*)d_ws;
  // workspace layout (floats)
  const size_t o_sum   = 0;
  const size_t o_memin = 16;                        // 1000*257
  const size_t o_th    = o_memin + 257024;          // 1000*256 (rounded slot)
  const size_t o_temp  = o_th    + 256000;          // 1000*256
  const size_t o_h1    = o_temp  + 256000;          // 50000*256
  const size_t o_es1   = o_h1    + 12800000;        // 50000*8
  const size_t o_ed1   = o_es1   + 400000;
  const size_t o_m1    = o_ed1   + 400000;
  const size_t o_z1    = o_m1    + 400000;
  const size_t o_w1    = o_z1    + 400000;          // 850000*8
  const size_t o_g1out = o_w1    + 6800000;         // 50000*256
  const size_t o_h2    = o_g1out + 12800000;        // 50000*256
  const size_t o_es2   = o_h2    + 12800000;        // 50000
  const size_t o_ed2   = o_es2   + 50000;
  const size_t o_m2    = o_ed2   + 50000;
  const size_t o_z2    = o_m2    + 50000;
  const size_t o_w2    = o_z2    + 50000;           // 850000
  const size_t o_chid  = o_h1;                      // reuse: h1 dead after layer-1 agg

  float* outF   = (float*)d_out;
  float* gat2   = outF;                             // [50000,256] first output
  float* logits = outF + (size_t)N_NODES * HID;     // [50000,2]

  dim3 blk(256);
  dim3 gemm_grid(HID / 64, (N_NODES + 63) / 64);    // N=256 for every big GEMM
  dim3 gemm_grid_mem(HID / 64, (MEM_N + 63) / 64);

  // ---- temporal embedding ----
  fill_f32<<<1, 1, 0, stream>>>(ws + o_sum, 0.f, 1);
  reduce_sum<<<64, 256, 0, stream>>>(ts, N_NODES, ws + o_sum);
  build_memin<<<(MEM_N * 257 + 255) / 256, blk, 0, stream>>>(node_mem, mem_time,
                                                             ws + o_sum, ws + o_memin);
  gemm_wmma_f32<<<gemm_grid_mem, blk, 0, stream>>>(ws + o_memin, te_w1, te_b1,
                                                   ws + o_th, MEM_N, HID, 257, /*relu*/1);
  gemm_wmma_f32<<<gemm_grid_mem, blk, 0, stream>>>(ws + o_th, te_w2, te_b2,
                                                   ws + o_temp, MEM_N, HID, HID, 0);

  // ---- h1 = concat(x, temporal) @ g1_w, without materializing xt ----
  gemm_wmma_f32<<<gemm_grid, blk, 0, stream>>>(x, g1_w, nullptr,
                                               ws + o_h1, N_NODES, HID, HID, 0);
  gemm_wmma_f32<<<gemm_grid_mem, blk, 0, stream>>>(ws + o_temp, g1_w + 256 * HID, nullptr,
                                                   ws + o_h1, MEM_N, HID, HID, /*accum*/2);

  // ---- GAT layer 1 (H=8, F=32) ----
  attn_scores<<<(N_NODES * 8 + 255) / 256, blk, 0, stream>>>(
      ws + o_h1, g1_asrc, g1_adst, ws + o_es1, ws + o_ed1, N_NODES, 8, 32);
  fill_f32<<<512, 256, 0, stream>>>(ws + o_m1, -INFINITY, (long long)N_NODES * 8);
  fill_f32<<<512, 256, 0, stream>>>(ws + o_z1, 0.f, (long long)N_NODES * 8);
  edge_max<<<((long long)E_TOT * 8 + 255) / 256, blk, 0, stream>>>(
      ei, ws + o_es1, ws + o_ed1, ws + o_w1, ws + o_m1, 8);
  edge_expsum<<<((long long)E_TOT * 8 + 255) / 256, blk, 0, stream>>>(
      ei, ws + o_w1, ws + o_m1, ws + o_z1, 8);
  init_bias_rows<<<2048, 256, 0, stream>>>(ws + o_g1out, g1_b, N_NODES);
  edge_aggregate<<<E_TOT, blk, 0, stream>>>(ei, ws + o_h1, ws + o_w1, ws + o_z1,
                                            ws + o_g1out, 8);
  relu_ip<<<2048, 256, 0, stream>>>(ws + o_g1out, (long long)N_NODES * HID);

  // ---- GAT layer 2 (H=1, F=256) -> writes into d_out ----
  gemm_wmma_f32<<<gemm_grid, blk, 0, stream>>>(ws + o_g1out, g2_w, nullptr,
                                               ws + o_h2, N_NODES, HID, HID, 0);
  attn_scores<<<(N_NODES + 255) / 256, blk, 0, stream>>>(
      ws + o_h2, g2_asrc, g2_adst, ws + o_es2, ws + o_ed2, N_NODES, 1, 256);
  fill_f32<<<256, 256, 0, stream>>>(ws + o_m2, -INFINITY, N_NODES);
  fill_f32<<<256, 256, 0, stream>>>(ws + o_z2, 0.f, N_NODES);
  edge_max<<<(E_TOT + 255) / 256, blk, 0, stream>>>(
      ei, ws + o_es2, ws + o_ed2, ws + o_w2, ws + o_m2, 1);
  edge_expsum<<<(E_TOT + 255) / 256, blk, 0, stream>>>(
      ei, ws + o_w2, ws + o_m2, ws + o_z2, 1);
  init_bias_rows<<<2048, 256, 0, stream>>>(gat2, g2_b, N_NODES);
  edge_aggregate<<<E_TOT, blk, 0, stream>>>(ei, ws + o_h2, ws + o_w2, ws + o_z2,
                                            gat2, 1);

  // ---- classifier ----
  gemm_wmma_f32<<<gemm_grid, blk, 0, stream>>>(gat2, c_w1, c_b1,
                                               ws + o_chid, N_NODES, HID, HID, /*relu*/1);
  logits_kernel<<<(N_NODES + 7) / 8, blk, 0, stream>>>(ws + o_chid, c_w2, c_b2,
                                                       logits, N_NODES);
}